// GraphDAU_8796093022332
// MI455X (gfx1250) — compile-verified
//
#include <hip/hip_runtime.h>

#define BB 4
#define NN 4096
#define EE 16384
#define CC 3
#define LL 4
#define BEC (BB * EE * CC)   // 196608
#define NJ  (NN * 16)        // 65536, padded node matrix [N][16]

typedef __attribute__((ext_vector_type(2))) float v2f;
typedef __attribute__((ext_vector_type(8))) float v8f;

// ---------------------------------------------------------------------------
// 1) Extract heads/tails from dense incidence matrix M [E, N] (+1 head, -1 tail)
// ---------------------------------------------------------------------------
__global__ void __launch_bounds__(256) extract_edges_kernel(
    const float* __restrict__ M, int* __restrict__ heads, int* __restrict__ tails) {
  int e = blockIdx.x * blockDim.x + threadIdx.x;
  if (e >= EE) return;
  const float4* row = (const float4*)(M + (size_t)e * NN);
  int h = 0, t = 0;
  bool hf = false, tf = false;
  for (int i = 0; i < NN / 4; ++i) {
    float4 v = row[i];
    if (v.x > 0.5f) { h = 4 * i + 0; hf = true; } else if (v.x < -0.5f) { t = 4 * i + 0; tf = true; }
    if (v.y > 0.5f) { h = 4 * i + 1; hf = true; } else if (v.y < -0.5f) { t = 4 * i + 1; tf = true; }
    if (v.z > 0.5f) { h = 4 * i + 2; hf = true; } else if (v.z < -0.5f) { t = 4 * i + 2; tf = true; }
    if (v.w > 0.5f) { h = 4 * i + 3; hf = true; } else if (v.w < -0.5f) { t = 4 * i + 3; tf = true; }
    if (hf && tf) break;
  }
  heads[e] = h;
  tails[e] = t;
}

// ---------------------------------------------------------------------------
// 2) Copy z0/y0 into mutable workspace (inputs must not be modified)
// ---------------------------------------------------------------------------
__global__ void __launch_bounds__(256) init_zy_kernel(
    const float* __restrict__ z0, const float* __restrict__ y0,
    float* __restrict__ z, float* __restrict__ y) {
  int i = blockIdx.x * blockDim.x + threadIdx.x;
  if (i >= BEC) return;
  z[i] = z0[i];
  y[i] = y0[i];
}

__global__ void __launch_bounds__(256) zero_kernel(float* __restrict__ p, int n) {
  int i = blockIdx.x * blockDim.x + threadIdx.x;
  if (i < n) p[i] = 0.0f;
}

// ---------------------------------------------------------------------------
// 3) b = M^T (z - y): scatter over edges into node accumulator [N][16], j=b*3+c
// ---------------------------------------------------------------------------
__global__ void __launch_bounds__(256) scatter_b_kernel(
    const float* __restrict__ z, const float* __restrict__ y,
    const int* __restrict__ heads, const int* __restrict__ tails,
    float* __restrict__ bacc) {
  int tid = blockIdx.x * blockDim.x + threadIdx.x;
  if (tid >= EE * 12) return;
  int e = tid / 12;
  int j = tid % 12;
  int b = j / 3, c = j % 3;
  int idx = (b * EE + e) * CC + c;
  float val = z[idx] - y[idx];
  atomicAdd(&bacc[heads[e] * 16 + j], val);
  atomicAdd(&bacc[tails[e] * 16 + j], -val);
}

// ---------------------------------------------------------------------------
// 4) Gl[n][j] = g[b,n,c] + bacc[n][j]/gamma_c ; columns 12..15 zero-padded
// ---------------------------------------------------------------------------
__global__ void __launch_bounds__(256) compute_gl_kernel(
    const float* __restrict__ g, const float* __restrict__ bacc,
    const float* __restrict__ gammas, int layer, float* __restrict__ Gl) {
  int tid = blockIdx.x * blockDim.x + threadIdx.x;
  if (tid >= NJ) return;
  int nrow = tid >> 4;
  int j = tid & 15;
  float outv = 0.0f;
  if (j < 12) {
    int b = j / 3, c = j % 3;
    float gamma = gammas[layer * CC + c];
    outv = g[((size_t)b * NN + nrow) * CC + c] + bacc[tid] / gamma;
  }
  Gl[tid] = outv;
}

// ---------------------------------------------------------------------------
// 5) t = U^T * Gl  (A[m,k] = U[k,m], direct coalesced loads) with fused
//    spectral scaling t[m,j] *= gamma_c/(gamma_c + eigenV[m]).
//    One 16-row M tile per block; 4 waves split K; deterministic LDS reduce.
// ---------------------------------------------------------------------------
__global__ void __launch_bounds__(128) gemm_t_kernel(
    const float* __restrict__ U, const float* __restrict__ Gl,
    const float* __restrict__ eigenV, const float* __restrict__ gammas,
    int layer, float* __restrict__ tout) {
  __shared__ float red[4][32][8];
  const int lane = threadIdx.x & 31;
  const int wave = threadIdx.x >> 5;
  const int m0 = blockIdx.x * 16;
  const int half = lane >> 4;   // 0: K 0,1   1: K 2,3
  const int n = lane & 15;      // B/C column, A row-in-tile
  const int kbeg = wave * (NN / 4);
  const int kend = kbeg + (NN / 4);

  v8f acc = {};
#pragma unroll 8
  for (int k0 = kbeg; k0 < kend; k0 += 4) {
    const int kr = k0 + half * 2;
    v2f a, bf;
    a.x = U[(size_t)kr * NN + m0 + n];          // A[m,k]   = U[k, m0+m]
    a.y = U[(size_t)(kr + 1) * NN + m0 + n];
    bf.x = Gl[kr * 16 + n];                      // B[k,j]   = Gl[k, j]
    bf.y = Gl[(kr + 1) * 16 + n];
    acc = __builtin_amdgcn_wmma_f32_16x16x4_f32(false, a, false, bf,
                                                (short)0, acc, false, false);
  }
#pragma unroll
  for (int v = 0; v < 8; ++v) red[wave][lane][v] = acc[v];
  __syncthreads();
  if (wave == 0) {
    const int c = n % 3;
    const float gamma = gammas[layer * CC + c];
#pragma unroll
    for (int v = 0; v < 8; ++v) {
      float s = red[0][lane][v] + red[1][lane][v] + red[2][lane][v] + red[3][lane][v];
      int m = m0 + v + 8 * half;                 // C/D layout: M = v + 8*(lane>=16)
      float ev = eigenV[m];
      tout[m * 16 + n] = s * (gamma / (gamma + ev));
    }
  }
}

// ---------------------------------------------------------------------------
// 6) x = U * t  (A[m,k] = U[m,k], transposed lane pattern -> LDS staging).
//    16x64 fp32 A tiles staged via ds_store_b128; pitch 68 -> conflict-free
//    ds_load_b64 A-fragment reads. 4-wave split-K + LDS reduce, as above.
// ---------------------------------------------------------------------------
__global__ void __launch_bounds__(128) gemm_x_kernel(
    const float* __restrict__ U, const float* __restrict__ tin,
    float* __restrict__ xout) {
  __shared__ float As[4][16][68];
  __shared__ float red[4][32][8];
  const int lane = threadIdx.x & 31;
  const int wave = threadIdx.x >> 5;
  const int i0 = blockIdx.x * 16;
  const int half = lane >> 4;
  const int n = lane & 15;
  const int kbeg = wave * (NN / 4);

  v8f acc = {};
  for (int kc = kbeg; kc < kbeg + NN / 4; kc += 64) {
    // Stage A tile: rows i0..i0+15, cols kc..kc+63 (coalesced 128b loads)
#pragma unroll
    for (int it = 0; it < 8; ++it) {
      int linear = it * 32 + lane;   // 0..255
      int row = linear >> 4;         // 0..15
      int c4 = linear & 15;          // 0..15 (float4 column)
      float4 vv = *(const float4*)&U[(size_t)(i0 + row) * NN + kc + c4 * 4];
      *(float4*)&As[wave][row][c4 * 4] = vv;
    }
    __syncthreads();   // uniform trip count across all 4 waves
#pragma unroll
    for (int kk = 0; kk < 64; kk += 4) {
      const int col = kk + half * 2;
      v2f a = *(const v2f*)&As[wave][n][col];    // A[m,k] = U[i0+m, k]
      const int k = kc + col;
      v2f bf;
      bf.x = tin[k * 16 + n];                    // B[k,j] = t_scaled[k, j]
      bf.y = tin[(k + 1) * 16 + n];
      acc = __builtin_amdgcn_wmma_f32_16x16x4_f32(false, a, false, bf,
                                                  (short)0, acc, false, false);
    }
    __syncthreads();
  }
#pragma unroll
  for (int v = 0; v < 8; ++v) red[wave][lane][v] = acc[v];
  __syncthreads();
  if (wave == 0) {
#pragma unroll
    for (int v = 0; v < 8; ++v) {
      float s = red[0][lane][v] + red[1][lane][v] + red[2][lane][v] + red[3][lane][v];
      int m = i0 + v + 8 * half;
      xout[m * 16 + n] = s;
    }
  }
}

// ---------------------------------------------------------------------------
// 7) Edge update: Mx = x[head]-x[tail]; z = soft_thr(Mx + y, beta); y = s - z
// ---------------------------------------------------------------------------
__global__ void __launch_bounds__(256) edge_update_kernel(
    const float* __restrict__ x, const int* __restrict__ heads,
    const int* __restrict__ tails, const float* __restrict__ betas,
    int layer, float* __restrict__ z, float* __restrict__ y) {
  int tid = blockIdx.x * blockDim.x + threadIdx.x;
  if (tid >= EE * 12) return;
  int e = tid / 12;
  int j = tid % 12;
  int b = j / 3, c = j % 3;
  float Mx = x[heads[e] * 16 + j] - x[tails[e] * 16 + j];
  int idx = (b * EE + e) * CC + c;
  float s = Mx + y[idx];
  float beta = betas[layer * CC + c];
  float a = fabsf(s) - beta;
  a = a > 0.0f ? a : 0.0f;
  float zn = (s >= 0.0f) ? a : -a;   // sign(0)*... == 0 since a<=0 then
  z[idx] = zn;
  y[idx] = s - zn;
}

// ---------------------------------------------------------------------------
// 8) Repack x [N][16] -> out [B, N, C]
// ---------------------------------------------------------------------------
__global__ void __launch_bounds__(256) write_out_kernel(
    const float* __restrict__ x, float* __restrict__ out) {
  int tid = blockIdx.x * blockDim.x + threadIdx.x;
  if (tid >= BB * NN * CC) return;
  int b = tid / (NN * CC);
  int rem = tid % (NN * CC);
  int nrow = rem / CC;
  int c = rem % CC;
  out[tid] = x[nrow * 16 + b * CC + c];
}

// ---------------------------------------------------------------------------
extern "C" void kernel_launch(void* const* d_in, const int* in_sizes, int n_in,
                              void* d_out, int out_size, void* d_ws, size_t ws_size,
                              hipStream_t stream) {
  const float* g      = (const float*)d_in[0];  // [B,N,C]
  const float* z0     = (const float*)d_in[1];  // [B,E,C]
  const float* y0     = (const float*)d_in[2];  // [B,E,C]
  const float* U      = (const float*)d_in[3];  // [N,N]
  const float* eigenV = (const float*)d_in[4];  // [N]
  const float* M      = (const float*)d_in[5];  // [E,N]
  const float* gammas = (const float*)d_in[6];  // [L,C]
  const float* betas  = (const float*)d_in[7];  // [L,C]
  float* out = (float*)d_out;

  char* ws = (char*)d_ws;
  size_t off = 0;
  int*   heads = (int*)(ws + off);   off += (size_t)EE * 4;
  int*   tails = (int*)(ws + off);   off += (size_t)EE * 4;
  float* zb    = (float*)(ws + off); off += (size_t)BEC * 4;
  float* yb    = (float*)(ws + off); off += (size_t)BEC * 4;
  float* bacc  = (float*)(ws + off); off += (size_t)NJ * 4;
  float* Gl    = (float*)(ws + off); off += (size_t)NJ * 4;
  float* tbuf  = (float*)(ws + off); off += (size_t)NJ * 4;
  float* xbuf  = (float*)(ws + off); off += (size_t)NJ * 4;

  extract_edges_kernel<<<(EE + 255) / 256, 256, 0, stream>>>(M, heads, tails);
  init_zy_kernel<<<(BEC + 255) / 256, 256, 0, stream>>>(z0, y0, zb, yb);

  for (int layer = 0; layer < LL; ++layer) {
    zero_kernel<<<(NJ + 255) / 256, 256, 0, stream>>>(bacc, NJ);
    scatter_b_kernel<<<(EE * 12 + 255) / 256, 256, 0, stream>>>(zb, yb, heads, tails, bacc);
    compute_gl_kernel<<<(NJ + 255) / 256, 256, 0, stream>>>(g, bacc, gammas, layer, Gl);
    gemm_t_kernel<<<NN / 16, 128, 0, stream>>>(U, Gl, eigenV, gammas, layer, tbuf);
    gemm_x_kernel<<<NN / 16, 128, 0, stream>>>(U, tbuf, xbuf);
    edge_update_kernel<<<(EE * 12 + 255) / 256, 256, 0, stream>>>(xbuf, heads, tails, betas,
                                                                  layer, zb, yb);
  }
  write_out_kernel<<<(BB * NN * CC + 255) / 256, 256, 0, stream>>>(xbuf, out);
}